// SplatPushModel_68994354643551
// MI455X (gfx1250) — compile-verified
//
#include <hip/hip_runtime.h>
#include <stdint.h>

// ---------------------------------------------------------------------------
// SplatPushModel for MI455X (gfx1250, wave32)
//   Stage 1: base = rho*(1-mask)               (plain coalesced stores)
//   Stage 2: bilinear scatter of rho*mask      (global_atomic_add_f32, guarded)
//   Stage 3: 10 redistribution iterations fully in LDS per 64x64 tile
//            (constant-shift splat == 4-tap gather stencil, 20px halo).
//            Tile fill: TDM tensor_load_to_lds (interior tiles, TENSORcnt)
//                       or async global->LDS B128 (edge tiles, ASYNCcnt).
//            Tile drain: async LDS->global B128.
// Everything stays in occ's native layout: rho(y,x) <-> idx = b*N*N + x*N + y,
// which absorbs both transposes in the reference.
// d_ws must hold B*N*N floats (64 MiB) for the intermediate rho_new.
// ---------------------------------------------------------------------------

#define Bn    64
#define Nn    512
#define TILE  64
#define HALO  20              // 10 iters * max 2px/iter source offset
#define LT    (TILE + 2*HALO) // 104 (multiple of 4 -> B128 alignment works)
#define ITERS 10

// ---------------- CDNA5 async global<->LDS helpers -------------------------
typedef int v4i __attribute__((__vector_size__(16)));
typedef __attribute__((address_space(1))) v4i* gv4p;
typedef __attribute__((address_space(3))) v4i* lv4p;

// generic->AS1: value-preserving; generic->AS3: low 32 bits == LDS offset
// (LLVM lowers generic->local addrspacecast as a truncate).
__device__ __forceinline__ gv4p to_global_v4(const float* p) {
  return (gv4p)(uintptr_t)p;
}
__device__ __forceinline__ lv4p to_lds_v4(const float* p) {
  return (lv4p)(uint32_t)(uintptr_t)p;
}

__device__ __forceinline__ void async_g2l_b128(float* lds, const float* g) {
#if __has_builtin(__builtin_amdgcn_global_load_async_to_lds_b128)
  __builtin_amdgcn_global_load_async_to_lds_b128(to_global_v4(g), to_lds_v4(lds), 0, 0);
#else
  unsigned l32 = (unsigned)(uintptr_t)lds;
  asm volatile("global_load_async_to_lds_b128 %0, %1, off"
               :: "v"(l32), "v"(g) : "memory");
#endif
}

__device__ __forceinline__ void async_l2g_b128(float* g, const float* lds) {
#if __has_builtin(__builtin_amdgcn_global_store_async_from_lds_b128)
  __builtin_amdgcn_global_store_async_from_lds_b128(to_global_v4(g), to_lds_v4(lds), 0, 0);
#else
  unsigned l32 = (unsigned)(uintptr_t)lds;
  asm volatile("global_store_async_from_lds_b128 %0, %1, off"
               :: "v"(l32), "v"(g) : "memory");
#endif
}

__device__ __forceinline__ void wait_async0() {
#if __has_builtin(__builtin_amdgcn_s_wait_asynccnt)
  __builtin_amdgcn_s_wait_asynccnt(0);
#else
  asm volatile("s_wait_asynccnt 0" ::: "memory");
#endif
}

// ---------------- CDNA5 TDM (Tensor Data Mover) ----------------------------
#if __has_builtin(__builtin_amdgcn_tensor_load_to_lds)
#define HAVE_TDM 1
typedef unsigned int v4u  __attribute__((ext_vector_type(4)));
typedef int          v8ie __attribute__((ext_vector_type(8)));
typedef int          v4ie __attribute__((ext_vector_type(4)));

// One-shot 2D tile DMA: LT x LT floats from A[gx0..][gy0..] (row stride Nn)
// into contiguous LDS at lds_byte_off. D# packing per CDNA5 ISA ch.8.
// This toolchain's builtin is the 6-arg form: (g0, g1, g2, g3, <v8i>, cpol);
// the trailing v8i is passed zero-filled (count=1 marks one valid D#).
__device__ __forceinline__ void tdm_load_tile(uint64_t gaddr, uint32_t lds_byte_off) {
  v4u g0;
  g0.x = 1u;                                           // count=1, user D#
  g0.y = lds_byte_off;                                 // lds_addr
  g0.z = (uint32_t)gaddr;                              // global_addr[31:0]
  g0.w = ((uint32_t)(gaddr >> 32) & 0x01FFFFFFu)       // global_addr[56:32]
         | 0x80000000u;                                // type=2 ("image")
  v8ie g1;
  g1[0] = 0x00020000;                                  // data_size=4B, no mcast
  g1[1] = (int)((unsigned)Nn << 16);                   // tensor_dim0 lo16
  g1[2] = (int)((unsigned)Nn << 16);                   // dim0 hi=0 | tensor_dim1 lo16
  g1[3] = (int)((unsigned)LT << 16);                   // dim1 hi=0 | tile_dim0
  g1[4] = LT;                                          // tile_dim1 | tile_dim2=0
  g1[5] = Nn;                                          // tensor_dim0_stride lo32
  g1[6] = 0;                                           // stride hi | dim1_stride lo
  g1[7] = 0;                                           // dim1_stride hi
  v4ie g2 = {1, 0, 0, 0};                              // tensor_dim2=1 (unused)
  v4ie g3 = {0, 0, 0, 0};
  v8ie g4 = {0, 0, 0, 0, 0, 0, 0, 0};                  // zero-filled trailing arg
  __builtin_amdgcn_tensor_load_to_lds(g0, g1, g2, g3, g4, 0);
}

__device__ __forceinline__ void wait_tensor0() {
#if __has_builtin(__builtin_amdgcn_s_wait_tensorcnt)
  __builtin_amdgcn_s_wait_tensorcnt(0);
#else
  asm volatile("s_wait_tensorcnt 0" ::: "memory");
#endif
}
#endif // HAVE_TDM

// ---------------- push geometry --------------------------------------------
__device__ __forceinline__ float sigf(float z) {
  return 1.0f / (1.0f + __expf(-z));
}

struct Geom { float L, ux, uy, s, mask; };

__device__ __forceinline__ Geom push_geom(float x, float y,
                                          float p0x, float p0y,
                                          float p1x, float p1y) {
  Geom gm;
  float dx = p1x - p0x, dy = p1y - p0y;
  gm.L = sqrtf(dx*dx + dy*dy + 1e-12f);
  float inv = 1.0f / gm.L;
  gm.ux = dx * inv;  gm.uy = dy * inv;
  float rx = x - p0x, ry = y - p0y;
  gm.s = rx*gm.ux + ry*gm.uy;
  float t = ry*gm.ux - rx*gm.uy;      // n = (-uy, ux)
  gm.mask = sigf(3.0f - fabsf(t)) * sigf(gm.s) * sigf(gm.L - gm.s);
  return gm;
}

// ---------------- Stage 1: base = rho*(1-mask) ------------------------------
__global__ void push_base_k(const float* __restrict__ occ,
                            const float* __restrict__ p0,
                            const float* __restrict__ p1,
                            float* __restrict__ A) {
  int idx = blockIdx.x * blockDim.x + threadIdx.x;
  int y = idx & (Nn - 1);
  int x = (idx >> 9) & (Nn - 1);
  int b = idx >> 18;
  Geom gm = push_geom((float)x, (float)y,
                      p0[2*b], p0[2*b+1], p1[2*b], p1[2*b+1]);
  A[idx] = occ[idx] * (1.0f - gm.mask);
}

// ---------------- Stage 2: bilinear scatter of moved mass -------------------
__global__ void push_scatter_k(const float* __restrict__ occ,
                               const float* __restrict__ p0,
                               const float* __restrict__ p1,
                               float* __restrict__ A) {
  int idx = blockIdx.x * blockDim.x + threadIdx.x;
  int y = idx & (Nn - 1);
  int x = (idx >> 9) & (Nn - 1);
  int b = idx >> 18;
  Geom gm = push_geom((float)x, (float)y,
                      p0[2*b], p0[2*b+1], p1[2*b], p1[2*b+1]);
  float moved = occ[idx] * gm.mask;
  float disp = fmaxf(gm.L - gm.s, 0.0f);
  float xf = (float)x + disp * gm.ux;
  float yf = (float)y + disp * gm.uy;
  float x0 = floorf(xf), y0 = floorf(yf);
  float fx = xf - x0,    fy = yf - y0;
  int xi = (int)x0, yi = (int)y0;
  size_t base = (size_t)b * (size_t)(Nn * Nn);
#pragma unroll
  for (int k = 0; k < 4; ++k) {
    int ddx = k & 1, ddy = k >> 1;
    int xc = xi + ddx, yc = yi + ddy;
    float w = (ddx ? fx : 1.0f - fx) * (ddy ? fy : 1.0f - fy);
    float contrib = w * moved;
    // sigmoid tails underflow to exact 0 far from the band: skipping
    // zero-contributions is bit-identical (base >= +0) and kills ~95% atomics.
    if (contrib != 0.0f && xc >= 0 && xc < Nn && yc >= 0 && yc < Nn)
      atomicAdd(&A[base + (size_t)xc * Nn + yc], contrib);
  }
}

// ---------------- Stage 3: 10 redistribution iters in LDS -------------------
// grid: (64 tiles, B); block: 256; dynamic LDS: 2*LT*LT floats (86.5 KB)
__global__ void redist_k(const float* __restrict__ A,
                         const float* __restrict__ p0,
                         const float* __restrict__ p1,
                         float* __restrict__ out) {
  extern __shared__ float smem[];
  float* bufA = smem;
  float* bufB = smem + LT * LT;

  const int b   = blockIdx.y;
  const int tx0 = (blockIdx.x & 7) * TILE;
  const int ty0 = (blockIdx.x >> 3) * TILE;
  const int tid = threadIdx.x;
  const size_t gbase = (size_t)b * (size_t)(Nn * Nn);

  float dxv = p1[2*b]   - p0[2*b];
  float dyv = p1[2*b+1] - p0[2*b+1];
  float dn  = sqrtf(dxv*dxv + dyv*dyv);
  bool  doRed = dn > 1e-6f;                 // uniform per block
  float inv = 1.0f / fmaxf(dn, 1e-6f);
  float ux = dxv * inv, uy = dyv * inv;

  // halo tile fully inside the grid? (block-uniform)
  const bool interior = (tx0 >= HALO) && (ty0 >= HALO) &&
                        (tx0 - HALO + LT <= Nn) && (ty0 - HALO + LT <= Nn);

  // ---- fill (TILE+2*HALO)^2 tile ----
#ifdef HAVE_TDM
  if (interior) {
    if (tid < 32) {   // tensor ops issue per-wave (EXEC ignored): wave 0 only
      uint64_t gaddr = (uint64_t)(uintptr_t)
          &A[gbase + (size_t)(tx0 - HALO) * Nn + (ty0 - HALO)];
      tdm_load_tile(gaddr, (uint32_t)(uintptr_t)bufA);
      wait_tensor0();
    }
  } else
#endif
  {
    const int NQ = LT * (LT / 4);           // vec4 slots, y-contiguous
    for (int q = tid; q < NQ; q += blockDim.x) {
      int ly = (q % (LT / 4)) * 4;
      int lx = q / (LT / 4);
      int gx = tx0 - HALO + lx;
      int gy = ty0 - HALO + ly;             // multiple of 4; clips are too
      float* lslot = &bufA[lx * LT + ly];
      if (gx >= 0 && gx < Nn && gy >= 0 && gy < Nn) {
        async_g2l_b128(lslot, &A[gbase + (size_t)gx * Nn + gy]);
      } else {
        lslot[0] = 0.0f; lslot[1] = 0.0f; lslot[2] = 0.0f; lslot[3] = 0.0f;
      }
    }
    wait_async0();
  }
  __syncthreads();

  float* res = bufA;
  if (doRed) {
    // constant-shift splat == 4-tap gather with fixed offsets/weights
    int   cx = (int)floorf(ux), cy = (int)floorf(uy);
    float fx = ux - (float)cx,  fy = uy - (float)cy;
    int   ox[4], oy[4]; float w[4];
#pragma unroll
    for (int k = 0; k < 4; ++k) {
      int ddx = k & 1, ddy = k >> 1;
      ox[k] = cx + ddx; oy[k] = cy + ddy;
      w[k]  = (ddx ? fx : 1.0f - fx) * (ddy ? fy : 1.0f - fy);
    }
    float* src = bufA; float* dst = bufB;
    for (int it = 0; it < ITERS; ++it) {
      for (int e = tid; e < LT * LT; e += blockDim.x) {
        int ly = e % LT;
        int lx = e / LT;
        float r   = src[e];
        float acc = r - fmaxf(r - 1.0f, 0.0f);     // r - exc
#pragma unroll
        for (int k = 0; k < 4; ++k) {
          int slx = lx - ox[k], sly = ly - oy[k];
          if (slx >= 0 && slx < LT && sly >= 0 && sly < LT)
            acc += w[k] * fmaxf(src[slx * LT + sly] - 1.0f, 0.0f);
        }
        // out-of-grid cells must stay 0 (reference drops OOB scatter mass);
        // can only trigger on edge tiles.
        if (!interior) {
          int gx = tx0 - HALO + lx, gy = ty0 - HALO + ly;
          if (gx < 0 || gx >= Nn || gy < 0 || gy >= Nn) acc = 0.0f;
        }
        dst[e] = acc;
      }
      __syncthreads();
      float* t = src; src = dst; dst = t;
    }
    res = src;   // 10 swaps -> back in bufA
  }

  // ---- async drain of 64x64 interior (always fully in-grid) ----
  const int NQo = TILE * (TILE / 4);
  for (int q = tid; q < NQo; q += blockDim.x) {
    int iy = (q % (TILE / 4)) * 4;
    int ix = q / (TILE / 4);
    async_l2g_b128(&out[gbase + (size_t)(tx0 + ix) * Nn + (ty0 + iy)],
                   &res[(HALO + ix) * LT + (HALO + iy)]);
  }
  wait_async0();
}

// ---------------------------------------------------------------------------
extern "C" void kernel_launch(void* const* d_in, const int* in_sizes, int n_in,
                              void* d_out, int out_size, void* d_ws, size_t ws_size,
                              hipStream_t stream) {
  const float* occ = (const float*)d_in[0];
  const float* p0  = (const float*)d_in[1];
  const float* p1  = (const float*)d_in[2];
  float* out = (float*)d_out;
  float* A   = (float*)d_ws;                 // needs Bn*Nn*Nn*4 = 64 MiB

  const int total = Bn * Nn * Nn;
  const int blk = 256;

  push_base_k   <<<total / blk, blk, 0, stream>>>(occ, p0, p1, A);
  push_scatter_k<<<total / blk, blk, 0, stream>>>(occ, p0, p1, A);

  const size_t ldsBytes = (size_t)2 * LT * LT * sizeof(float); // 86528 B
  (void)hipFuncSetAttribute((const void*)redist_k,
                            hipFuncAttributeMaxDynamicSharedMemorySize,
                            (int)ldsBytes);
  redist_k<<<dim3(64, Bn), blk, ldsBytes, stream>>>(A, p0, p1, out);
}